// MonotonicAttention_43662637532056
// MI455X (gfx1250) — compile-verified
//
#include <hip/hip_runtime.h>
#include <hip/hip_bf16.h>

// MonotonicAttention for MI455X (gfx1250, wave32).
// Phases: (0) w = g*wv/||wv||  (1) p[b,s] from tanh-dot energy (reads k, 256MB)
//         (2) linear-recurrence scan -> alpha  (3) zero ctx  (4) ctx = alpha@v via
//         V_WMMA_F32_16X16X4_F32 with replicated-alpha A operand (reads v, 256MB).

typedef float v2f __attribute__((ext_vector_type(2)));
typedef float v8f __attribute__((ext_vector_type(8)));

#define NTHR 256

// ---------------- Kernel 0: weight-normalized projection vector ----------------
__global__ void wnorm_kernel(const float* __restrict__ wv, const float* __restrict__ g,
                             float* __restrict__ w, int D) {
    __shared__ float red[NTHR];
    int t = threadIdx.x;
    float s = 0.f;
    for (int d = t; d < D; d += NTHR) { float x = wv[d]; s += x * x; }
    red[t] = s;
    __syncthreads();
    for (int off = NTHR / 2; off > 0; off >>= 1) {
        if (t < off) red[t] += red[t + off];
        __syncthreads();
    }
    float scale = g[0] * rsqrtf(red[0]);
    for (int d = t; d < D; d += NTHR) w[d] = scale * wv[d];
}

// ---------------- Kernel 1: selection probability p[b,s] ----------------
// block = 256 threads = 8 waves; wave handles one s-row; grid = (S/8, B).
__global__ void energy_kernel(const float* __restrict__ q, const float* __restrict__ k,
                              const float* __restrict__ noise, const int* __restrict__ k_len,
                              const float* __restrict__ bvec, const float* __restrict__ w,
                              const float* __restrict__ bias, const float* __restrict__ r,
                              float* __restrict__ p_out, int S, int D) {
    __shared__ float sqb[1024];  // q[b,:] + bvec
    __shared__ float sw[1024];   // w
    int b = blockIdx.y;
    for (int d = threadIdx.x; d < D; d += NTHR) {
        sqb[d] = q[(size_t)b * D + d] + bvec[d];
        sw[d]  = w[d];
    }
    __syncthreads();

    int wave = threadIdx.x >> 5;
    int lane = threadIdx.x & 31;
    int s = blockIdx.x * 8 + wave;
    const float* __restrict__ krow = k + ((size_t)b * S + s) * D;

    float acc = 0.f;
#pragma unroll
    for (int i = 0; i < 8; ++i) {
        int d = i * 128 + lane * 4;
        __builtin_prefetch(krow + d + 512, 0, 0);  // global_prefetch_b8, stream ahead
        float4 kv = *(const float4*)(krow + d);
        acc += tanhf(sqb[d + 0] + kv.x) * sw[d + 0];
        acc += tanhf(sqb[d + 1] + kv.y) * sw[d + 1];
        acc += tanhf(sqb[d + 2] + kv.z) * sw[d + 2];
        acc += tanhf(sqb[d + 3] + kv.w) * sw[d + 3];
    }
    // wave32 butterfly reduction
#pragma unroll
    for (int off = 16; off >= 1; off >>= 1) acc += __shfl_xor(acc, off, 32);

    if (lane == 0) {
        int kl = k_len[b];
        float pv;
        if (s >= kl) {
            pv = 0.f;                     // masked: sigmoid(-inf + noise) = 0
        } else if (s == kl - 1) {
            pv = 1.f;                     // end_mask: force selection
        } else {
            float e = acc + bias[0] + r[0] + noise[(size_t)b * S + s];
            pv = 1.f / (1.f + expf(-e));
        }
        p_out[(size_t)b * S + s] = pv;
    }
}

// ---------------- Kernel 2: recurrence scan  a_j = m_j*a_{j-1} + x_j ----------------
// One block per batch; 256 threads x 8 elems; pair-composition Hillis-Steele scan.
__global__ void scan_kernel(const float* __restrict__ p, const float* __restrict__ prev_att,
                            float* __restrict__ alpha_ws, float* __restrict__ alpha_out,
                            int S) {
    const int E = 8;  // S = 2048 = 256 * 8
    int b = blockIdx.x;
    int t = threadIdx.x;
    int s0 = t * E;

    float m[E], x[E], pv[E];
#pragma unroll
    for (int i = 0; i < E; ++i) {
        int s = s0 + i;
        pv[i] = p[(size_t)b * S + s];
        m[i]  = (s == 0) ? 1.f : (1.f - p[(size_t)b * S + s - 1]);
        x[i]  = prev_att[(size_t)b * S + s];
    }
    // local composed transform T(a) = M*a + A  (applied left-to-right)
    float M = 1.f, A = 0.f;
#pragma unroll
    for (int i = 0; i < E; ++i) { A = m[i] * A + x[i]; M = m[i] * M; }

    __shared__ float sM[NTHR], sA[NTHR];
    sM[t] = M; sA[t] = A;
    __syncthreads();
    for (int off = 1; off < NTHR; off <<= 1) {
        float cm = sM[t], ca = sA[t];
        float pm = 1.f, pa = 0.f;
        if (t >= off) { pm = sM[t - off]; pa = sA[t - off]; }
        __syncthreads();
        sM[t] = cm * pm;            // later ∘ earlier
        sA[t] = cm * pa + ca;
        __syncthreads();
    }
    float a = (t == 0) ? 0.f : sA[t - 1];  // exclusive prefix applied to init 0
#pragma unroll
    for (int i = 0; i < E; ++i) {
        a = m[i] * a + x[i];
        float al = pv[i] * a;
        alpha_ws[(size_t)b * S + s0 + i]  = al;
        alpha_out[(size_t)b * S + s0 + i] = al;
    }
}

// ---------------- Kernel 3: zero ctx accumulator ----------------
__global__ void zero_kernel(float* __restrict__ ptr, int n) {
    int i = blockIdx.x * blockDim.x + threadIdx.x;
    if (i < n) ptr[i] = 0.f;
}

// ---------------- Kernel 4: ctx = alpha @ v via WMMA f32 16x16x4 ----------------
// Replicated-A trick: A[m,k] = alpha[s0+k] (all m), B[k,n] = v[s0+k, d0+n]
// => D[m,n] = sum_k alpha*v, identical across m. Accumulate in C over the s-range.
// f32 16x16x4 layout (wave32): A/B in 2 VGPRs, upper half-wave holds K offset +2:
//   reg0 = K=2*khalf, reg1 = K=2*khalf+1, with khalf = lane>>4, n = lane&15.
// block = 128 threads = 4 waves; wave owns a 16-wide d tile.
// grid = (D/64, S/256, B); s-splits reduced with global f32 atomics.
__global__ void ctx_wmma_kernel(const float* __restrict__ alpha, const float* __restrict__ v,
                                float* __restrict__ ctx, int S, int D) {
    int b     = blockIdx.z;
    int wave  = threadIdx.x >> 5;
    int lane  = threadIdx.x & 31;
    int dbase = blockIdx.x * 64 + wave * 16;
    int sbase = blockIdx.y * 256;
    int khalf = lane >> 4;
    int n     = lane & 15;

    const float* __restrict__ arow = alpha + (size_t)b * S + sbase;
    const float* __restrict__ vrow = v + ((size_t)b * S + sbase) * D;

    v8f c = {0.f, 0.f, 0.f, 0.f, 0.f, 0.f, 0.f, 0.f};
#pragma unroll 4
    for (int it = 0; it < 64; ++it) {
        int klo = it * 4 + 2 * khalf;
        v2f A, B;
        A.x = arow[klo];
        A.y = arow[klo + 1];
        const float* vp = vrow + (size_t)klo * D + dbase + n;
        B.x = vp[0];
        B.y = vp[D];
        __builtin_prefetch(vp + 16 * (size_t)D, 0, 0);
        // (neg_a, A, neg_b, B, c_mod, C, reuse_a, reuse_b)
        c = __builtin_amdgcn_wmma_f32_16x16x4_f32(false, A, false, B, (short)0, c,
                                                  false, false);
    }
    // Row M=0 of the (replicated) result lives in C-vgpr0, lanes 0..15.
    if (lane < 16) atomicAdd(&ctx[(size_t)b * D + dbase + n], c[0]);
}

extern "C" void kernel_launch(void* const* d_in, const int* in_sizes, int n_in,
                              void* d_out, int out_size, void* d_ws, size_t ws_size,
                              hipStream_t stream) {
    const float* q        = (const float*)d_in[0];
    const float* k        = (const float*)d_in[1];
    const float* v        = (const float*)d_in[2];
    const float* prev_att = (const float*)d_in[3];
    const float* noise    = (const float*)d_in[4];
    const int*   k_len    = (const int*)d_in[5];
    const float* bvec     = (const float*)d_in[6];
    const float* wv       = (const float*)d_in[7];
    const float* g        = (const float*)d_in[8];
    const float* bias     = (const float*)d_in[9];
    const float* r        = (const float*)d_in[10];

    const int B = in_sizes[5];             // k_len: [B]
    const int D = in_sizes[6];             // b: [D]
    const int S = in_sizes[4] / B;         // noise: [B,S]

    float* out      = (float*)d_out;       // [B*D ctx][B*S alpha]
    float* ctx_out  = out;
    float* alpha_out = out + (size_t)B * D;

    float* ws       = (float*)d_ws;
    float* w_ws     = ws;                       // D
    float* p_ws     = ws + D;                   // B*S
    float* alpha_ws = ws + D + (size_t)B * S;   // B*S

    // 0: weight-normalized direction
    wnorm_kernel<<<1, NTHR, 0, stream>>>(wv, g, w_ws, D);

    // 1: selection probabilities (streams k, 256 MB)
    energy_kernel<<<dim3(S / 8, B), NTHR, 0, stream>>>(q, k, noise, k_len, bvec, w_ws,
                                                       bias, r, p_ws, S, D);

    // 2: recurrence scan -> alpha (also written to output slot)
    scan_kernel<<<B, NTHR, 0, stream>>>(p_ws, prev_att, alpha_ws, alpha_out, S);

    // 3: zero ctx accumulator (atomics target)
    zero_kernel<<<(B * D + NTHR - 1) / NTHR, NTHR, 0, stream>>>(ctx_out, B * D);

    // 4: ctx = alpha @ v with WMMA (streams v, 256 MB)
    ctx_wmma_kernel<<<dim3(D / 64, S / 256, B), 128, 0, stream>>>(alpha_ws, v, ctx_out,
                                                                  S, D);
}